// NodeEncoder_71622874628185
// MI455X (gfx1250) — compile-verified
//
#include <hip/hip_runtime.h>
#include <hip/hip_bf16.h>

typedef __attribute__((ext_vector_type(2))) float v2f;
typedef __attribute__((ext_vector_type(8))) float v8f;

#define N_NODES 100000
#define EMB_DIM 128
#define N_EDGES 600000
#define KTOT    256             // 2*EMB_DIM (concat[x, x_upd])
#define KT4     (KTOT/4)        // 64 K-tiles of 4
#define NT16    (EMB_DIM/16)    // 8 N-tiles of 16
#define WPK_ELEMS (KT4*NT16*32) // v2f elements in packed W (16384 = 128KB)
#define WPK_HALF  (WPK_ELEMS/2) // 8192 v2f = 64KB per K-phase

// ---------------- degree kernels ----------------
__global__ void k_deg_init(float* __restrict__ deg_s, float* __restrict__ deg_r) {
    int i = blockIdx.x * blockDim.x + threadIdx.x;
    if (i < N_NODES) { deg_s[i] = 1.0f; deg_r[i] = 1.0f; } // self-loop baked in
}

__global__ void k_deg_edges(const int* __restrict__ senders, const int* __restrict__ receivers,
                            float* __restrict__ deg_s, float* __restrict__ deg_r) {
    int e = blockIdx.x * blockDim.x + threadIdx.x;
    if (e < N_EDGES) {
        atomicAdd(&deg_s[senders[e]],   1.0f);
        atomicAdd(&deg_r[receivers[e]], 1.0f);
    }
}

// in-place: deg_s -> rsqrt(deg_s), deg_r -> deg_r^{-3/2}  (deg >= 1 always)
__global__ void k_coef(float* __restrict__ deg_s, float* __restrict__ deg_r) {
    int i = blockIdx.x * blockDim.x + threadIdx.x;
    if (i < N_NODES) {
        float ds = deg_s[i];
        float dr = deg_r[i];
        deg_s[i] = rsqrtf(ds);
        deg_r[i] = rsqrtf(dr) / dr;
    }
}

// ---------------- pack W (256x128 row-major) into WMMA B-fragment order ----------------
// B 4x16 f32 layout (mirrors C/D striping): VGPR0: lanes0-15 = K=k0 row, lanes16-31 = K=k0+2;
// VGPR1: K=k0+1 / K=k0+3.  Pack so the wave fragment is one contiguous b64 load per lane.
__global__ void k_pack_w(const float* __restrict__ W, v2f* __restrict__ Wpk) {
    int i = blockIdx.x * blockDim.x + threadIdx.x;     // kt*256 + nt*32 + lane
    if (i >= WPK_ELEMS) return;
    int lane = i & 31;
    int nt   = (i >> 5) & (NT16 - 1);
    int kt   = i >> 8;
    int half = lane >> 4, ln = lane & 15;
    int k = kt * 4 + 2 * half;
    int n = nt * 16 + ln;
    v2f w;
    w.x = W[(size_t)k       * EMB_DIM + n];
    w.y = W[(size_t)(k + 1) * EMB_DIM + n];
    Wpk[i] = w;
}

// ---------------- summed init: self-loop contribution summed[i] = x[i]*rsqrt(deg_s[i]) ----
__global__ void k_init_summed(const float* __restrict__ x, const int* __restrict__ gid,
                              const float* __restrict__ inv_s, float* __restrict__ summed) {
    int t = blockIdx.x * blockDim.x + threadIdx.x;     // over N_NODES * (EMB_DIM/4)
    if (t >= N_NODES * (EMB_DIM / 4)) return;
    int node = t / (EMB_DIM / 4);
    int c4   = t % (EMB_DIM / 4);
    int row  = gid ? gid[node] : node;
    const float4 xv = *(const float4*)(x + (size_t)row * EMB_DIM + c4 * 4);
    float is = inv_s[node];
    float4 o = make_float4(xv.x * is, xv.y * is, xv.z * is, xv.w * is);
    *(float4*)(summed + (size_t)node * EMB_DIM + c4 * 4) = o;
}

// ---------------- edge scatter: one wave per edge, float4 per lane, fp32 atomics ----------
__global__ __launch_bounds__(256)
void k_scatter(const float* __restrict__ x, const int* __restrict__ gid,
               const float* __restrict__ inv_s,
               const int* __restrict__ senders, const int* __restrict__ receivers,
               float* __restrict__ summed) {
    int w    = (blockIdx.x * blockDim.x + threadIdx.x) >> 5;  // edge id (wave-uniform)
    int lane = threadIdx.x & 31;
    if (w >= N_EDGES) return;
    int s = senders[w];
    int r = receivers[w];
    int srow = gid ? gid[s] : s;
    float is = inv_s[s];
    const float4 xv = *(const float4*)(x + (size_t)srow * EMB_DIM + lane * 4);
    float* dst = summed + (size_t)r * EMB_DIM + lane * 4;
    atomicAdd(dst + 0, xv.x * is);
    atomicAdd(dst + 1, xv.y * is);
    atomicAdd(dst + 2, xv.z * is);
    atomicAdd(dst + 3, xv.w * is);
}

// ---------------- fused GEMM: out = [x | summed*coef_r] @ W + b, optional ReLU -----------
// One wave owns a 16-row x 128-col output tile. K = 256 in 64 WMMA f32 16x16x4 steps.
// W is staged through LDS in two 64KB phases so each workgroup reads W from L2 exactly
// once (8x less L2 traffic than per-wave global reads); B fragments come from ds_load_b64
// (lane -> consecutive 8B slots: conflict-free across 64 banks).
__global__ __launch_bounds__(256)
void k_gemm(const float* __restrict__ x, const int* __restrict__ gid,
            const float* __restrict__ summed, const float* __restrict__ coef_r,
            const v2f* __restrict__ Wpk, const float* __restrict__ bias,
            float* __restrict__ out, int do_relu) {
    __shared__ v2f Wlds[WPK_HALF];                     // 64KB stage (of 320KB WGP LDS)

    int wave  = (blockIdx.x * blockDim.x + threadIdx.x) >> 5;
    int lane  = threadIdx.x & 31;
    bool valid = wave < (N_NODES / 16);                // 100000 = 16*6250, wave-uniform
    int rows0 = wave * 16;
    int half  = lane >> 4, ln = lane & 15;
    int m     = valid ? (rows0 + ln) : 0;              // A-row this lane feeds
    int xrow  = gid ? gid[m] : m;
    const float* arow = x      + (size_t)xrow * EMB_DIM;
    const float* srow = summed + (size_t)m    * EMB_DIM;
    float cr = coef_r[m];

    v8f acc[NT16];
#pragma unroll
    for (int nt = 0; nt < NT16; ++nt)
#pragma unroll
        for (int r = 0; r < 8; ++r) acc[nt][r] = 0.0f;

    // ---- stage phase-1 weights (K rows 0..127 -> ktiles 0..31), 64KB ----
    {
        const float4* src = (const float4*)(Wpk);
        float4*       dst = (float4*)Wlds;
        for (int i = threadIdx.x; i < WPK_HALF / 2; i += 256) dst[i] = src[i];
    }
    __syncthreads();

    if (valid) {
        // K-phase 1: original features x (unscaled)
        for (int k0 = 0; k0 < EMB_DIM; k0 += 4) {
            v2f a = *(const v2f*)(arow + k0 + 2 * half);  // contiguous K pair per lane
            const v2f* bp = Wlds + ((k0 >> 2) * NT16) * 32 + lane;
#pragma unroll
            for (int nt = 0; nt < NT16; ++nt) {
                v2f b = bp[nt * 32];
                acc[nt] = __builtin_amdgcn_wmma_f32_16x16x4_f32(
                    false, a, false, b, (short)0, acc[nt], false, false);
            }
        }
    }
    __syncthreads();                                   // all waves done reading phase 1

    // ---- stage phase-2 weights (K rows 128..255 -> ktiles 32..63), 64KB ----
    {
        const float4* src = (const float4*)(Wpk + WPK_HALF);
        float4*       dst = (float4*)Wlds;
        for (int i = threadIdx.x; i < WPK_HALF / 2; i += 256) dst[i] = src[i];
    }
    __syncthreads();

    if (valid) {
        // K-phase 2: aggregated neighbors, degree scale folded into A load
        for (int k0 = 0; k0 < EMB_DIM; k0 += 4) {
            v2f a = *(const v2f*)(srow + k0 + 2 * half);
            a.x *= cr; a.y *= cr;
            const v2f* bp = Wlds + ((k0 >> 2) * NT16) * 32 + lane;
#pragma unroll
            for (int nt = 0; nt < NT16; ++nt) {
                v2f b = bp[nt * 32];
                acc[nt] = __builtin_amdgcn_wmma_f32_16x16x4_f32(
                    false, a, false, b, (short)0, acc[nt], false, false);
            }
        }
        // epilogue: bias (+ReLU), store per C/D layout: VGPR r -> row rows0 + r + 8*half
#pragma unroll
        for (int nt = 0; nt < NT16; ++nt) {
            float bv = bias[nt * 16 + ln];
#pragma unroll
            for (int r = 0; r < 8; ++r) {
                float v = acc[nt][r] + bv;
                if (do_relu) v = fmaxf(v, 0.0f);
                out[(size_t)(rows0 + r + 8 * half) * EMB_DIM + nt * 16 + ln] = v;
            }
        }
    }
}

extern "C" void kernel_launch(void* const* d_in, const int* in_sizes, int n_in,
                              void* d_out, int out_size, void* d_ws, size_t ws_size,
                              hipStream_t stream) {
    (void)in_sizes; (void)n_in; (void)out_size; (void)ws_size;
    const int*   gid       = (const int*)d_in[0];
    const int*   senders   = (const int*)d_in[1];
    const int*   receivers = (const int*)d_in[2];
    // d_in[3] = is_training (0) -> dropout is identity
    const float* emb = (const float*)d_in[4];
    const float* W1  = (const float*)d_in[5];
    const float* b1  = (const float*)d_in[6];
    const float* W2  = (const float*)d_in[7];
    const float* b2  = (const float*)d_in[8];
    float* out = (float*)d_out;

    float* ws     = (float*)d_ws;
    float* deg_s  = ws;                                   // -> rsqrt(deg_s)
    float* deg_r  = deg_s + N_NODES;                      // -> deg_r^{-3/2}
    float* summed = deg_r + N_NODES;                      // [N, 128]
    float* h1r    = summed + (size_t)N_NODES * EMB_DIM;   // [N, 128] relu(layer1)
    v2f*   Wpk1   = (v2f*)(h1r + (size_t)N_NODES * EMB_DIM);
    v2f*   Wpk2   = Wpk1 + WPK_ELEMS;

    const dim3 blk(256);
    const int gN   = (N_NODES + 255) / 256;
    const int gE   = (N_EDGES + 255) / 256;
    const int gPk  = (WPK_ELEMS + 255) / 256;
    const int gEl  = (N_NODES * (EMB_DIM / 4) + 255) / 256;    // 12500
    const int gSc  = (N_EDGES * 32 + 255) / 256;               // 75000
    const int gGm  = ((N_NODES / 16) * 32 + 255) / 256;        // 782

    // degrees + coefficients (shared by both layers)
    k_deg_init <<<gN,  blk, 0, stream>>>(deg_s, deg_r);
    k_deg_edges<<<gE,  blk, 0, stream>>>(senders, receivers, deg_s, deg_r);
    k_coef     <<<gN,  blk, 0, stream>>>(deg_s, deg_r);
    // pack weights into WMMA B-fragment order
    k_pack_w   <<<gPk, blk, 0, stream>>>(W1, Wpk1);
    k_pack_w   <<<gPk, blk, 0, stream>>>(W2, Wpk2);

    // ---- layer 1: x = emb_table[gid] ----
    k_init_summed<<<gEl, blk, 0, stream>>>(emb, gid, deg_s, summed);
    k_scatter    <<<gSc, blk, 0, stream>>>(emb, gid, deg_s, senders, receivers, summed);
    k_gemm       <<<gGm, blk, 0, stream>>>(emb, gid, summed, deg_r, Wpk1, b1, h1r, 1);

    // ---- layer 2: x = relu(h1) ----
    k_init_summed<<<gEl, blk, 0, stream>>>(h1r, nullptr, deg_s, summed);
    k_scatter    <<<gSc, blk, 0, stream>>>(h1r, nullptr, deg_s, senders, receivers, summed);
    k_gemm       <<<gGm, blk, 0, stream>>>(h1r, nullptr, summed, deg_r, Wpk2, b2, out, 0);
}